// Lookback_48404281426420
// MI455X (gfx1250) — compile-verified
//
#include <hip/hip_runtime.h>

typedef __attribute__((ext_vector_type(2))) float v2f;
typedef __attribute__((ext_vector_type(8))) float v8f;

// Problem constants (from reference: x is [8, 4096, 1024] fp32)
constexpr int B = 8, T = 4096, C = 1024;
constexpr int NSEG = 32, SEGLEN = 128;      // NSEG*SEGLEN == T
constexpr int CT1 = C / 256;                // 256-wide column tiles for passes 1/2

// ---------------- Pass 1: per-segment column sums -> ws ----------------
// ws[b, seg, c] = sum_{t in segment} x[b, t, c]        (1 MB total)
__global__ void __launch_bounds__(256) seg_sum_kernel(const float* __restrict__ x,
                                                      float* __restrict__ sums) {
    const int tid = threadIdx.x;
    const int blk = blockIdx.x;
    const int ct  = blk % CT1;
    const int seg = (blk / CT1) % NSEG;
    const int b   = blk / (CT1 * NSEG);
    const int c   = ct * 256 + tid;
    const float* xp = x + ((size_t)b * T + (size_t)seg * SEGLEN) * C + c;
    float s = 0.f;
#pragma unroll 8
    for (int t = 0; t < SEGLEN; ++t) s += xp[(size_t)t * C];
    sums[((size_t)b * NSEG + seg) * C + c] = s;
}

// ---------------- Pass 2: in-place exclusive scan over segments ----------------
__global__ void __launch_bounds__(256) seg_scan_kernel(float* __restrict__ sums) {
    const int tid = threadIdx.x;
    const int blk = blockIdx.x;
    const int ct  = blk % CT1;
    const int b   = blk / CT1;
    const int c   = ct * 256 + tid;
    float run = 0.f;
    for (int seg = 0; seg < NSEG; ++seg) {
        const size_t idx = ((size_t)b * NSEG + seg) * C + c;
        const float v = sums[idx];
        sums[idx] = run;
        run += v;
    }
}

// ---------------- Pass 3: WMMA triangular cumsum + 1/(t+1) scaling ----------------
// One wave handles 16 columns x one 128-long T segment.
// Per 16-row tile: D = L(16x16, lower-tri ones) * X(16x16) + carry, via 4x
// V_WMMA_F32_16X16X4_F32 accumulations; carry chains through row 15.
__global__ void __launch_bounds__(256) cumsum_wmma_kernel(const float* __restrict__ x,
                                                          const float* __restrict__ pre,
                                                          float* __restrict__ out) {
    const int lane = threadIdx.x & 31;
    const int wid  = blockIdx.x * 8 + (threadIdx.x >> 5);
    const int ctile = wid % (C / 16);
    const int seg   = (wid / (C / 16)) % NSEG;
    const int b     = wid / ((C / 16) * NSEG);

    const int n = lane & 15;   // A: row M; B: col N; D: col N
    const int h = lane >> 4;   // half-wave select
    const int col = ctile * 16 + n;

    // Exclusive prefix of earlier segments for this column
    float carry = pre[((size_t)b * NSEG + seg) * C + col];

    const int tseg = seg * SEGLEN;
    for (int tile = 0; tile < SEGLEN / 16; ++tile) {
        const int t0 = tseg + tile * 16;
        const float* xp = x + ((size_t)b * T + t0) * C + col;

        if (tile + 1 < SEGLEN / 16)
            __builtin_prefetch(xp + (size_t)16 * C, 0, 1);

        v8f acc = {carry, carry, carry, carry, carry, carry, carry, carry};
#pragma unroll
        for (int q = 0; q < 4; ++q) {
            // B operand (4x16): VGPR r, half h -> K = r + 2h, i.e. t = t0 + 4q + r + 2h
            v2f bb;
            bb[0] = xp[(size_t)(4 * q + 0 + 2 * h) * C];
            bb[1] = xp[(size_t)(4 * q + 1 + 2 * h) * C];
            // A operand (16x4) = columns [4q, 4q+3] of lower-triangular ones:
            // A[m][k] = (m >= 4q + k), with m = lane&15, k = r + 2h
            v2f a;
            a[0] = (n >= 4 * q + 0 + 2 * h) ? 1.0f : 0.0f;
            a[1] = (n >= 4 * q + 1 + 2 * h) ? 1.0f : 0.0f;
            acc = __builtin_amdgcn_wmma_f32_16x16x4_f32(false, a, false, bb,
                                                        (short)0, acc, false, false);
        }

        // Next-tile carry = D row M=15 = acc[7] on lanes 16..31 (per column n)
        carry = __int_as_float(
            __builtin_amdgcn_ds_bpermute((16 + n) << 2, __float_as_int(acc[7])));

        // Store: VGPR m, half h -> t = t0 + m + 8h; scale by 1/(t+1).
        // Divisor is an exact small integer -> v_rcp_f32 + one FMA Newton step
        // is ~correctly rounded (<=1 ulp vs exact divide) at 3 VALU ops instead
        // of the ~10-op precise-division sequence.
        float* op = out + ((size_t)b * T + t0) * C + col;
#pragma unroll
        for (int m = 0; m < 8; ++m) {
            const int t = t0 + m + 8 * h;
            const float d = (float)(t + 1);
            float inv = __builtin_amdgcn_rcpf(d);
            inv = __builtin_fmaf(__builtin_fmaf(-d, inv, 1.0f), inv, inv);
            op[(size_t)(m + 8 * h) * C] = acc[m] * inv;
        }
    }
}

extern "C" void kernel_launch(void* const* d_in, const int* in_sizes, int n_in,
                              void* d_out, int out_size, void* d_ws, size_t ws_size,
                              hipStream_t stream) {
    (void)in_sizes; (void)n_in; (void)out_size; (void)ws_size;
    const float* x = (const float*)d_in[0];
    float* out = (float*)d_out;
    float* ws  = (float*)d_ws;   // needs B*NSEG*C*4 = 1 MB

    dim3 blk(256);
    seg_sum_kernel<<<B * NSEG * CT1, blk, 0, stream>>>(x, ws);
    seg_scan_kernel<<<B * CT1, blk, 0, stream>>>(ws);
    cumsum_wmma_kernel<<<(B * NSEG * (C / 16)) / 8, blk, 0, stream>>>(x, ws, out);
}